// GNN_Transfomer_29025388987166
// MI455X (gfx1250) — compile-verified
//
#include <hip/hip_runtime.h>
#include <math.h>

// Problem constants (from reference)
#define Bc   2
#define Tc   96
#define BT   (Bc * Tc)      // 192
#define Nn   1000
#define Ee   16000
#define EH   16
#define EOUT 30
#define NOUT 3
#define NTILES (Ee / 16)    // 1000 tiles of 16 edges

#define THREADS 512
#define NWAVES  (THREADS / 32)

typedef __attribute__((ext_vector_type(2))) float v2f;
typedef __attribute__((ext_vector_type(8))) float v8f;

// fast sigmoid: v_exp_f32 + v_rcp_f32 (single TRANS ops, no IEEE div expansion)
__device__ __forceinline__ float sigmoidf_(float x) {
    return __builtin_amdgcn_rcpf(1.0f + __expf(-x));
}

// D = A(16x4 f32) * B(4x16 f32) + C(16x16 f32), wave32 WMMA
__device__ __forceinline__ v8f wmma4(v2f a, v2f b, v8f c) {
    return __builtin_amdgcn_wmma_f32_16x16x4_f32(
        /*neg_a=*/false, a, /*neg_b=*/false, b,
        /*c_mod=*/(short)0, c, /*reuse_a=*/false, /*reuse_b=*/false);
}

// ---------------------------------------------------------------------------
// Kernel 1: edge_attr column mean / ddof-1 std  ->  stats[4] = {m0,m1,rstd0,rstd1}
// ---------------------------------------------------------------------------
__global__ __launch_bounds__(256) void ea_stats_kernel(const float* __restrict__ ea,
                                                       float* __restrict__ stats) {
    __shared__ float sh[4][256];
    const int tid = threadIdx.x;
    float s0 = 0.f, s1 = 0.f, q0 = 0.f, q1 = 0.f;
    for (int e = tid; e < Ee; e += 256) {
        float a = ea[2 * e + 0];
        float b = ea[2 * e + 1];
        s0 += a; q0 += a * a;
        s1 += b; q1 += b * b;
    }
    sh[0][tid] = s0; sh[1][tid] = s1; sh[2][tid] = q0; sh[3][tid] = q1;
    __syncthreads();
    for (int off = 128; off > 0; off >>= 1) {
        if (tid < off) {
            sh[0][tid] += sh[0][tid + off];
            sh[1][tid] += sh[1][tid + off];
            sh[2][tid] += sh[2][tid + off];
            sh[3][tid] += sh[3][tid + off];
        }
        __syncthreads();
    }
    if (tid == 0) {
        float m0 = sh[0][0] / (float)Ee;
        float m1 = sh[1][0] / (float)Ee;
        float v0 = (sh[2][0] - sh[0][0] * m0) / (float)(Ee - 1);  // ddof=1
        float v1 = (sh[3][0] - sh[1][0] * m1) / (float)(Ee - 1);
        stats[0] = m0;
        stats[1] = m1;
        stats[2] = rsqrtf(v0);
        stats[3] = rsqrtf(v1);
    }
}

// ---------------------------------------------------------------------------
// Kernel 2: fused edge-MLP (WMMA) + LDS scatter-add + node-MLP.
// One block per (b*t) slice; agg[1000][30] lives entirely in LDS.
// ---------------------------------------------------------------------------
__global__ __launch_bounds__(THREADS) void gnn_fused_kernel(
    const float* __restrict__ x,          // (BT, N, 3)
    const float* __restrict__ edge_attr,  // (E, 2)
    const float* __restrict__ wind_mean,  // (2,)
    const float* __restrict__ wind_std,   // (2,)
    const float* __restrict__ We1,        // (9, 16)
    const float* __restrict__ be1,        // (16,)
    const float* __restrict__ We2,        // (16, 30)
    const float* __restrict__ be2,        // (30,)
    const float* __restrict__ Wn,         // (30, 3)
    const float* __restrict__ bn,         // (3,)
    const int*   __restrict__ eidx,       // (2, E) int32
    const float* __restrict__ stats,      // {m0,m1,rstd0,rstd1}
    float* __restrict__ out)              // (BT, N, 3)
{
    extern __shared__ float smem[];
    float* sAgg = smem;                // Nn * EOUT floats (120 KB)
    float* sH   = smem + Nn * EOUT;    // NWAVES * 256 floats (16 KB)

    const int bt   = blockIdx.x;
    const int tid  = threadIdx.x;
    const int lane = tid & 31;
    const int wave = tid >> 5;
    const int half = lane >> 4;   // 0: lanes 0-15, 1: lanes 16-31
    const int col  = lane & 15;

    const float* __restrict__ xbt = x + (size_t)bt * Nn * 3;

    // --- zero the LDS aggregation table ---
    for (int i = tid; i < Nn * EOUT; i += THREADS) sAgg[i] = 0.0f;
    __syncthreads();

    const float wm0 = wind_mean[0], wm1 = wind_mean[1];
    const float wsd0 = wind_std[0], wsd1 = wind_std[1];
    const float m0 = stats[0], m1 = stats[1], r0 = stats[2], r1 = stats[3];

    // --- preload B operands in WMMA B-layout: lane holds B[k][n],
    //     n = lane%16, k = v + 2*(lane>=16) (+4*kstep) ---
    v2f B1[3];
#pragma unroll
    for (int kk = 0; kk < 3; ++kk) {
        int k0 = 4 * kk + 2 * half;
        B1[kk].x = (k0     < 9) ? We1[(k0)     * EH + col] : 0.0f;
        B1[kk].y = (k0 + 1 < 9) ? We1[(k0 + 1) * EH + col] : 0.0f;
    }
    const int n1 = col + 16;           // second N-tile column (16..31), valid if < 30
    v2f B2a[4], B2b[4];
#pragma unroll
    for (int kk = 0; kk < 4; ++kk) {
        int k0 = 4 * kk + 2 * half;    // always < 16
        B2a[kk].x = We2[(k0)     * EOUT + col];
        B2a[kk].y = We2[(k0 + 1) * EOUT + col];
        B2b[kk].x = (n1 < EOUT) ? We2[(k0)     * EOUT + n1] : 0.0f;
        B2b[kk].y = (n1 < EOUT) ? We2[(k0 + 1) * EOUT + n1] : 0.0f;
    }
    const float be1n  = be1[col];
    const float be2n0 = be2[col];
    const float be2n1 = (n1 < EOUT) ? be2[n1] : 0.0f;

    float* sHw = sH + wave * 256;      // per-wave 16x16 staging tile

    // --- edge tiles: each wave owns tiles wave, wave+NWAVES, ... (uniform per wave) ---
    for (int tile = wave; tile < NTILES; tile += NWAVES) {
        const int e    = tile * 16 + col;        // lanes 16-31 mirror lanes 0-15
        const int sIdx = eidx[e];                // row 0: src
        const int tIdx = eidx[Ee + e];           // row 1: tgt

        const float f0 = xbt[sIdx * 3 + 0];
        const float f1 = xbt[sIdx * 3 + 1];
        const float f2 = xbt[sIdx * 3 + 2];
        const float g0 = xbt[tIdx * 3 + 0];
        const float g1 = xbt[tIdx * 3 + 1];
        const float g2 = xbt[tIdx * 3 + 2];
        const float dist  = edge_attr[2 * e + 0];
        const float direc = edge_attr[2 * e + 1];

        const float wspd = f1 * wsd0 + wm0;
        const float wdir = f2 * wsd1 + wm1;
        // relu(3*wspd*cos(theta)/dist): v_cos + v_rcp (dist in [0.5,1.5], no div expansion)
        const float ew   = fmaxf(0.0f, 3.0f * wspd * __cosf(direc - wdir) *
                                        __builtin_amdgcn_rcpf(dist));
        const float ean0 = (dist  - m0) * r0;
        const float ean1 = (direc - m1) * r1;
        // feat[9] = {f0,f1,f2,g0,g1,g2,ean0,ean1,ew}, K padded to 12 with zeros

        // A-layout: lane holds A[m][k], m = lane%16, k = v + 2*(lane>=16) (+4*kstep)
        v2f a0, a1, a2;
        if (half) { a0.x = f2;   a0.y = g0;   a1.x = ean0; a1.y = ean1; a2.x = 0.0f; a2.y = 0.0f; }
        else      { a0.x = f0;   a0.y = f1;   a1.x = g1;   a1.y = g2;   a2.x = ew;   a2.y = 0.0f; }

        // GEMM1: (16x12) @ (12x16) -> 16x16
        v8f acc = {0.f,0.f,0.f,0.f,0.f,0.f,0.f,0.f};
        acc = wmma4(a0, B1[0], acc);
        acc = wmma4(a1, B1[1], acc);
        acc = wmma4(a2, B1[2], acc);

        // h = sigmoid(acc + be1); C-layout elem (m,n): vgpr r, lane n+16*(m>=8), m=r+8*half
        // stage to LDS as sHw[m*16 + k]
#pragma unroll
        for (int rr = 0; rr < 8; ++rr) {
            float h = sigmoidf_(acc[rr] + be1n);
            sHw[(rr + 8 * half) * 16 + col] = h;
        }
        asm volatile("" ::: "memory");  // keep DS store->load order (LDS is in-order per wave)

        // reload h in A-layout for GEMM2
        const float* hrow = sHw + col * 16;
        v2f A2[4];
#pragma unroll
        for (int kk = 0; kk < 4; ++kk) {
            A2[kk].x = hrow[4 * kk + 2 * half + 0];
            A2[kk].y = hrow[4 * kk + 2 * half + 1];
        }

        // GEMM2: (16x16) @ (16x30) -> 16x30 in two 16x16 N-tiles
        v8f accA = {0.f,0.f,0.f,0.f,0.f,0.f,0.f,0.f};
        v8f accB = {0.f,0.f,0.f,0.f,0.f,0.f,0.f,0.f};
#pragma unroll
        for (int kk = 0; kk < 4; ++kk) accA = wmma4(A2[kk], B2a[kk], accA);
#pragma unroll
        for (int kk = 0; kk < 4; ++kk) accB = wmma4(A2[kk], B2b[kk], accB);

        // scatter: row m = rr + 8*half needs src/tgt of edge tile*16+m.
        // lane m already holds them -> broadcast with ds_bpermute.
#pragma unroll
        for (int rr = 0; rr < 8; ++rr) {
            const int mrow = rr + 8 * half;
            const int s_r = __builtin_amdgcn_ds_bpermute(4 * mrow, sIdx);
            const int t_r = __builtin_amdgcn_ds_bpermute(4 * mrow, tIdx);
            const float v0 = sigmoidf_(accA[rr] + be2n0);
            atomicAdd(&sAgg[t_r * EOUT + col],  v0);
            atomicAdd(&sAgg[s_r * EOUT + col], -v0);
            if (n1 < EOUT) {
                const float v1 = sigmoidf_(accB[rr] + be2n1);
                atomicAdd(&sAgg[t_r * EOUT + n1],  v1);
                atomicAdd(&sAgg[s_r * EOUT + n1], -v1);
            }
        }
    }

    __syncthreads();

    // --- node MLP: out = sigmoid(agg @ Wn + bn) ---
    const float bn0 = bn[0], bn1 = bn[1], bn2 = bn[2];
    for (int n = tid; n < Nn; n += THREADS) {
        const float* a = sAgg + n * EOUT;
        float o0 = bn0, o1 = bn1, o2 = bn2;
#pragma unroll
        for (int j = 0; j < EOUT; ++j) {
            const float av = a[j];
            o0 += av * Wn[j * 3 + 0];
            o1 += av * Wn[j * 3 + 1];
            o2 += av * Wn[j * 3 + 2];
        }
        const size_t o = ((size_t)bt * Nn + n) * 3;
        out[o + 0] = sigmoidf_(o0);
        out[o + 1] = sigmoidf_(o1);
        out[o + 2] = sigmoidf_(o2);
    }
}

// ---------------------------------------------------------------------------
extern "C" void kernel_launch(void* const* d_in, const int* in_sizes, int n_in,
                              void* d_out, int out_size, void* d_ws, size_t ws_size,
                              hipStream_t stream) {
    const float* x         = (const float*)d_in[0];
    const float* edge_attr = (const float*)d_in[1];
    const float* wind_mean = (const float*)d_in[2];
    const float* wind_std  = (const float*)d_in[3];
    const float* We1       = (const float*)d_in[4];
    const float* be1       = (const float*)d_in[5];
    const float* We2       = (const float*)d_in[6];
    const float* be2       = (const float*)d_in[7];
    const float* Wn        = (const float*)d_in[8];
    const float* bn        = (const float*)d_in[9];
    const int*   eidx      = (const int*)d_in[10];
    float*       out       = (float*)d_out;
    float*       stats     = (float*)d_ws;   // 4 floats

    ea_stats_kernel<<<1, 256, 0, stream>>>(edge_attr, stats);

    const size_t shmem = (size_t)(Nn * EOUT + NWAVES * 256) * sizeof(float);  // ~136 KB
    gnn_fused_kernel<<<BT, THREADS, shmem, stream>>>(
        x, edge_attr, wind_mean, wind_std,
        We1, be1, We2, be2, Wn, bn, eidx, stats, out);
}